// Bevformernet_45810121179588
// MI455X (gfx1250) — compile-verified
//
#include <hip/hip_runtime.h>
#include <hip/hip_bf16.h>
#include <cstddef>

// ---- problem constants (from reference) ----
constexpr int Bc = 2;       // batch
constexpr int Sc = 2;       // cameras
constexpr int Nc = 4096;    // queries
constexpr int Mc = 1580;    // spatial tokens (20*79)
constexpr int Cc = 128;     // channels
constexpr int Hh = 4;       // heads
constexpr int HD = 32;      // head dim
constexpr int Pp = 128;     // points per head
constexpr int Dd = 32;      // Z anchors
constexpr int HfI = 20;
constexpr int WfI = 79;
constexpr int BS = Bc * Sc; // 4

typedef float v2f __attribute__((ext_vector_type(2)));
typedef float v8f __attribute__((ext_vector_type(8)));

// V_WMMA_F32_16X16X4_F32: A = 16x4 f32 (2 VGPRs/lane), B = 4x16 f32, C/D = 16x16 f32 (8 VGPRs)
__device__ __forceinline__ v8f wmma4(v2f a, v2f b, v8f c) {
  return __builtin_amdgcn_wmma_f32_16x16x4_f32(
      /*neg_a=*/false, a, /*neg_b=*/false, b,
      /*c_mod=*/(short)0, c, /*reuse_a=*/false, /*reuse_b=*/false);
}

// ------------------------------------------------------------------
// Kernel 1: value projection  v[bs][m][c] = value[s][m][b][:] @ Wv + bv
// one wave per 16x16 output tile; K=128 stepped by 4 (32 WMMAs).
// Edge rows (m >= Mc) are handled by address clamping: a garbage A row
// only affects the same output row, which is skipped at store time.
// ------------------------------------------------------------------
__global__ __launch_bounds__(32)
void vproj_kernel(const float* __restrict__ value, const float* __restrict__ Wv,
                  const float* __restrict__ bv, float* __restrict__ vproj) {
  const int c0    = blockIdx.x * 16;   // 0..127
  const int m0    = blockIdx.y * 16;   // 0..1583 (edge guarded at store)
  const int bsIdx = blockIdx.z;        // 0..3
  const int b = bsIdx / Sc, s = bsIdx % Sc;
  const int lane = threadIdx.x;
  const int lh  = lane >> 4;           // 0/1 -> K half
  const int l16 = lane & 15;

  const int am  = m0 + l16;
  const int amc = (am < Mc) ? am : (Mc - 1);   // clamp: memory-safe, no divergence
  const float* arow = value + (((size_t)s * Mc + amc) * Bc + b) * Cc;

  v8f acc = {};
  for (int k0 = 0; k0 < Cc; k0 += 4) {
    const int ka = k0 + lh * 2;
    const v2f a = *(const v2f*)(arow + ka);      // global_load_b64
    v2f bb;
    bb.x = Wv[(size_t)ka * Cc + c0 + l16];
    bb.y = Wv[(size_t)(ka + 1) * Cc + c0 + l16];
    acc = wmma4(a, bb, acc);
  }
  const float bias = bv[c0 + l16];
#pragma unroll
  for (int r = 0; r < 8; ++r) {
    const int m = m0 + r + 8 * lh;
    if (m < Mc)
      vproj[((size_t)bsIdx * Mc + m) * Cc + c0 + l16] = acc[r] + bias;
  }
}

// ------------------------------------------------------------------
// Kernel 2: fused per-query block (16 queries, one bs index).
// 4 waves; wave h owns head h: attn GEMM -> softmax -> offset GEMM
// tiles consumed immediately by bilinear sampling of vproj.
// ------------------------------------------------------------------
__global__ __launch_bounds__(128)
void fused_kernel(const float* __restrict__ query, const float* __restrict__ qpos,
                  const float* __restrict__ refpts,
                  const float* __restrict__ Woff, const float* __restrict__ boff,
                  const float* __restrict__ Wattn, const float* __restrict__ battn,
                  const float* __restrict__ vproj, float* __restrict__ outws) {
  __shared__ float qs[16][Cc];           // 8 KB  : q + query_pos tile
  __shared__ float refsS[16][Dd][2];     // 4 KB  : reference points
  __shared__ float attnS[Hh][16][Pp];    // 32 KB : softmaxed weights
  __shared__ float tileS[Hh][16][16];    // 4 KB  : per-wave offset staging

  const int n0    = blockIdx.x * 16;
  const int bsIdx = blockIdx.y;
  const int b = bsIdx / Sc, s = bsIdx % Sc;
  const int tid  = threadIdx.x;
  const int h    = tid >> 5;     // wave id == head
  const int lane = tid & 31;
  const int lh   = lane >> 4;
  const int l16  = lane & 15;

  // stage q = query + query_pos
  for (int i = tid; i < 16 * Cc; i += 128) {
    const int r = i >> 7, c = i & 127;
    const size_t gi = ((size_t)b * Nc + n0 + r) * Cc + c;
    qs[r][c] = query[gi] + qpos[gi];
  }
  // stage reference points  refpts: (S,B,N,D,2)
  for (int i = tid; i < 16 * Dd * 2; i += 128) {
    const int r = i >> 6;
    const int rem = i & 63;
    const int d = rem >> 1, comp = rem & 1;
    refsS[r][d][comp] =
        refpts[((((size_t)s * Bc + b) * Nc + (n0 + r)) * Dd + d) * 2 + comp];
  }
  __syncthreads();

  // ---- phase 1: attention logits for head h (cols h*128 .. h*128+127) ----
  for (int t = 0; t < 8; ++t) {
    const int col0 = h * Pp + t * 16;
    v8f acc = {};
    for (int k0 = 0; k0 < Cc; k0 += 4) {
      const int ka = k0 + lh * 2;
      const v2f a = *(const v2f*)(&qs[l16][ka]);   // ds_load_b64
      v2f bb;
      bb.x = Wattn[(size_t)ka * (Hh * Pp) + col0 + l16];
      bb.y = Wattn[(size_t)(ka + 1) * (Hh * Pp) + col0 + l16];
      acc = wmma4(a, bb, acc);
    }
    const float bias = battn[col0 + l16];
#pragma unroll
    for (int r = 0; r < 8; ++r)
      attnS[h][r + 8 * lh][t * 16 + l16] = acc[r] + bias;
  }
  __syncthreads();

  // ---- softmax over P=128 per (head, query); lanes 0..15 own one query ----
  if (lane < 16) {
    float mx = -1e30f;
    for (int p = 0; p < Pp; ++p) mx = fmaxf(mx, attnS[h][lane][p]);
    float sum = 0.0f;
    for (int p = 0; p < Pp; ++p) {
      const float e = __expf(attnS[h][lane][p] - mx);
      attnS[h][lane][p] = e;
      sum += e;
    }
    const float inv = 1.0f / sum;
    for (int p = 0; p < Pp; ++p) attnS[h][lane][p] *= inv;
  }
  __syncthreads();

  // ---- phase 2: offset GEMM tiles + immediate bilinear sampling ----
  float acc[16];
#pragma unroll
  for (int q = 0; q < 16; ++q) acc[q] = 0.0f;
  // lane = channel within head
  const float* vbase = vproj + (size_t)bsIdx * Mc * Cc + h * HD + lane;

  for (int t = 0; t < 16; ++t) {                 // 16 tiles x 16 cols = 256 cols/head
    const int col0 = h * (Pp * 2) + t * 16;      // column into Woff (1024 wide)
    v8f oacc = {};
    for (int k0 = 0; k0 < Cc; k0 += 4) {
      const int ka = k0 + lh * 2;
      const v2f a = *(const v2f*)(&qs[l16][ka]);
      v2f bb;
      bb.x = Woff[(size_t)ka * (Hh * Pp * 2) + col0 + l16];
      bb.y = Woff[(size_t)(ka + 1) * (Hh * Pp * 2) + col0 + l16];
      oacc = wmma4(a, bb, oacc);
    }
    const float bias = boff[col0 + l16];
#pragma unroll
    for (int r = 0; r < 8; ++r)
      tileS[h][r + 8 * lh][l16] = oacc[r] + bias;
    // same-wave LDS ops are in-order: the reads below observe the stores above

#pragma unroll 1
    for (int q = 0; q < 16; ++q) {
#pragma unroll 1
      for (int pl = 0; pl < 8; ++pl) {
        const int p = t * 8 + pl;          // point index; p = pz*D + d
        const float ox = tileS[h][q][pl * 2];
        const float oy = tileS[h][q][pl * 2 + 1];
        const int d = p & (Dd - 1);
        // loc*Wf - 0.5 with loc = ref + off/Wf  ==  ref*Wf + off - 0.5
        const float x = refsS[q][d][0] * (float)WfI + ox - 0.5f;
        const float y = refsS[q][d][1] * (float)HfI + oy - 0.5f;
        const float w = attnS[h][q][p];
        const float fx0 = floorf(x), fy0 = floorf(y);
        const int x0 = (int)fx0, y0 = (int)fy0;
        const float fx = x - fx0, fy = y - fy0;
        const float w00 = (1.0f - fx) * (1.0f - fy) * w;
        const float w10 = fx * (1.0f - fy) * w;
        const float w01 = (1.0f - fx) * fy * w;
        const float w11 = fx * fy * w;
        if ((unsigned)x0 < (unsigned)WfI && (unsigned)y0 < (unsigned)HfI)
          acc[q] += w00 * vbase[(size_t)(y0 * WfI + x0) * Cc];
        if ((unsigned)(x0 + 1) < (unsigned)WfI && (unsigned)y0 < (unsigned)HfI)
          acc[q] += w10 * vbase[(size_t)(y0 * WfI + x0 + 1) * Cc];
        if ((unsigned)x0 < (unsigned)WfI && (unsigned)(y0 + 1) < (unsigned)HfI)
          acc[q] += w01 * vbase[(size_t)((y0 + 1) * WfI + x0) * Cc];
        if ((unsigned)(x0 + 1) < (unsigned)WfI && (unsigned)(y0 + 1) < (unsigned)HfI)
          acc[q] += w11 * vbase[(size_t)((y0 + 1) * WfI + x0 + 1) * Cc];
      }
    }
  }

#pragma unroll
  for (int q = 0; q < 16; ++q)
    outws[((size_t)bsIdx * Nc + n0 + q) * Cc + h * HD + lane] = acc[q];
}

// ------------------------------------------------------------------
// Kernel 3: camera combine (keep mask from batch-0, count from per-batch
// hits) + output projection GEMM + bias + residual.
// ------------------------------------------------------------------
__global__ __launch_bounds__(128)
void out_kernel(const float* __restrict__ query,
                const unsigned char* __restrict__ mask,   // bev_mask bytes
                const float* __restrict__ outws,
                const float* __restrict__ Wout, const float* __restrict__ bout,
                float* __restrict__ dout) {
  __shared__ float sl[16][Cc];       // slots tile
  __shared__ float scaleS[16][Sc];   // keep[s] / count per row

  const int g0  = blockIdx.x * 16;   // rows into (B*N)
  const int tid = threadIdx.x;
  const int wv = tid >> 5, lane = tid & 31, lh = lane >> 4, l16 = lane & 15;

  if (tid < 16) {
    const int gr = g0 + tid;
    const int b = gr / Nc, n = gr % Nc;
    float cnt = 0.0f;
    float kp[Sc];
    for (int s = 0; s < Sc; ++s) {
      int sk = 0, sh = 0;
      for (int d = 0; d < Dd; ++d) {
        sk += mask[(((size_t)s * Bc + 0) * Nc + n) * Dd + d];
        sh += mask[(((size_t)s * Bc + b) * Nc + n) * Dd + d];
      }
      kp[s] = (sk > 0) ? 1.0f : 0.0f;
      cnt += (sh > 0) ? 1.0f : 0.0f;
    }
    const float inv = 1.0f / fmaxf(cnt, 1.0f);
    for (int s = 0; s < Sc; ++s) scaleS[tid][s] = kp[s] * inv;
  }
  __syncthreads();

  for (int i = tid; i < 16 * Cc; i += 128) {
    const int r = i >> 7, c = i & 127;
    const int gr = g0 + r;
    const int b = gr / Nc, n = gr % Nc;
    float v = 0.0f;
    for (int s = 0; s < Sc; ++s)
      v += outws[((size_t)(b * Sc + s) * Nc + n) * Cc + c] * scaleS[r][s];
    sl[r][c] = v;
  }
  __syncthreads();

  // 8 column tiles of W_out; 4 waves x 2 tiles
  for (int tt = 0; tt < 2; ++tt) {
    const int c0 = (wv * 2 + tt) * 16;
    v8f acc = {};
    for (int k0 = 0; k0 < Cc; k0 += 4) {
      const int ka = k0 + lh * 2;
      const v2f a = *(const v2f*)(&sl[l16][ka]);   // ds_load_b64
      v2f bb;
      bb.x = Wout[(size_t)ka * Cc + c0 + l16];
      bb.y = Wout[(size_t)(ka + 1) * Cc + c0 + l16];
      acc = wmma4(a, bb, acc);
    }
    const float bias = bout[c0 + l16];
#pragma unroll
    for (int r = 0; r < 8; ++r) {
      const size_t gr = (size_t)g0 + r + 8 * lh;
      dout[gr * Cc + c0 + l16] = acc[r] + bias + query[gr * Cc + c0 + l16];
    }
  }
}

// ------------------------------------------------------------------
extern "C" void kernel_launch(void* const* d_in, const int* in_sizes, int n_in,
                              void* d_out, int out_size, void* d_ws, size_t ws_size,
                              hipStream_t stream) {
  (void)in_sizes; (void)n_in; (void)out_size; (void)ws_size;
  const float* query  = (const float*)d_in[0];
  // d_in[1] = key (unused by the math)
  const float* value  = (const float*)d_in[2];
  const float* qpos   = (const float*)d_in[3];
  const float* refpts = (const float*)d_in[4];
  const unsigned char* bmask = (const unsigned char*)d_in[5];
  // d_in[6] spatial_shapes, d_in[7] level_start_index: compile-time constants
  const float* Wv    = (const float*)d_in[8];
  const float* bv    = (const float*)d_in[9];
  const float* Woff  = (const float*)d_in[10];
  const float* boff  = (const float*)d_in[11];
  const float* Wattn = (const float*)d_in[12];
  const float* battn = (const float*)d_in[13];
  const float* Wout  = (const float*)d_in[14];
  const float* bout  = (const float*)d_in[15];

  float* ws    = (float*)d_ws;
  float* vproj = ws;                               // BS*Mc*Cc   = 808,960 f
  float* outws = ws + (size_t)BS * Mc * Cc;        // BS*Nc*Cc   = 2,097,152 f
  float* dout  = (float*)d_out;                    // (B,N,C)

  // K1: value projection — tiles: 8 col x ceil(1580/16)=99 row x 4 bs
  vproj_kernel<<<dim3(8, 99, 4), 32, 0, stream>>>(value, Wv, bv, vproj);
  // K2: fused deformable attention — 256 query-tiles x 4 (b,s)
  fused_kernel<<<dim3(Nc / 16, BS), 128, 0, stream>>>(
      query, qpos, refpts, Woff, boff, Wattn, battn, vproj, outws);
  // K3: combine + output projection — (B*N)/16 = 512 row tiles
  out_kernel<<<dim3((Bc * Nc) / 16), 128, 0, stream>>>(
      query, bmask, outws, Wout, bout, dout);
}